// PointNet_Plus_32959579029661
// MI455X (gfx1250) — compile-verified
//
#include <hip/hip_runtime.h>

typedef float v2f __attribute__((ext_vector_type(2)));
typedef float v8f __attribute__((ext_vector_type(8)));

#define BATCH 32

// ============================================================================
// Weight pre-pack: W (cout x cin, row-major) -> WMMA B-fragment order.
// Packed layout (v2f units): Bp[(nt*ktiles + kt)*32 + lane]
//   lane<16 : col n = nt*16+lane, K = {kt*4,   kt*4+1}
//   lane>=16: col n = nt*16+lane-16, K = {kt*4+2, kt*4+3}
// Zero-padded where K >= cin -> inner GEMM loop needs no guards/branches.
// ============================================================================
__global__ void pack_w_kernel(const float* __restrict__ W, int cin, int cout,
                              float* __restrict__ out) {
  int kpad = (cin + 3) & ~3;
  int ktiles = kpad >> 2;
  int total = (cout >> 4) * ktiles * 64;  // 32 lanes * 2 floats
  int idx = blockIdx.x * blockDim.x + threadIdx.x;
  if (idx >= total) return;
  int j = idx & 1;
  int lane = (idx >> 1) & 31;
  int t = idx >> 6;
  int kt = t % ktiles;
  int nt = t / ktiles;
  int n = nt * 16 + (lane & 15);
  int k = kt * 4 + ((lane >> 4) << 1) + j;
  out[idx] = (k < cin) ? W[(size_t)n * cin + k] : 0.0f;
}

// ============================================================================
// Fused MLP tile: one wave computes a 16-row stripe of
//   Y = relu((X * W^T [+ biasMat]) * s + b)
// X rows in LDS (even stride -> A frag is one aligned ds_load_b64);
// W pre-packed in global (one coalesced global_load_b64 per fragment).
// Dual N-tile accumulators (cout % 32 == 0) for MMA ILP.
// Uses V_WMMA_F32_16X16X4_F32 (full f32 precision, wave32).
// ============================================================================
__device__ __forceinline__ void mlp_layer_tile(
    const float* __restrict__ bufIn, int sIn,
    float* __restrict__ bufOut, int sOut,
    int m0, int ktiles, int cout,
    const float* __restrict__ Wp,
    const float* __restrict__ sc, const float* __restrict__ bi,
    const float* __restrict__ biasMat, int rowBase, int Rvalid,
    int lane) {
  int half = (lane >> 4) & 1;
  int l15 = lane & 15;
  int arow = m0 + l15;
  const float* aBase = bufIn + arow * sIn + half * 2;  // even offset, 8B aligned
  const v2f* WpV = (const v2f*)Wp;
  for (int np = 0; np * 32 < cout; ++np) {
    v8f acc0 = {}, acc1 = {};
    const v2f* bp0 = WpV + ((size_t)(2 * np) * ktiles) * 32 + lane;
    const v2f* bp1 = bp0 + (size_t)ktiles * 32;
#pragma unroll 2
    for (int kt = 0; kt < ktiles; ++kt) {
      v2f a = *(const v2f*)(aBase + kt * 4);
      v2f b0 = bp0[kt * 32];
      v2f b1 = bp1[kt * 32];
      acc0 = __builtin_amdgcn_wmma_f32_16x16x4_f32(
          false, a, false, b0, (short)0, acc0, false, false);
      acc1 = __builtin_amdgcn_wmma_f32_16x16x4_f32(
          false, a, false, b1, (short)0, acc1, false, false);
    }
    int col0 = np * 32 + l15;
    int col1 = col0 + 16;
    float sc0 = sc[col0], sc1 = sc[col1];
    float bi0 = bi[col0], bi1 = bi[col1];
#pragma unroll
    for (int i = 0; i < 8; ++i) {
      int row = m0 + i + half * 8;
      float v0 = acc0[i], v1 = acc1[i];
      if (biasMat != nullptr) {
        int gr = rowBase + row;
        if (gr < Rvalid) {
          v0 += biasMat[gr * cout + col0];
          v1 += biasMat[gr * cout + col1];
        }
      }
      v0 = fmaxf(v0 * sc0 + bi0, 0.0f);
      v1 = fmaxf(v1 * sc1 + bi1, 0.0f);
      bufOut[row * sOut + col0] = v0;
      bufOut[row * sOut + col1] = v1;
    }
  }
}

// ---------------- transpose (B,C,N) -> (B,N,C) ----------------
__global__ void transpose_bcn_kernel(const float* __restrict__ in,
                                     float* __restrict__ out, int C, int N) {
  int idx = blockIdx.x * blockDim.x + threadIdx.x;
  int total = BATCH * C * N;
  if (idx >= total) return;
  int n = idx % N;
  int c = (idx / N) % C;
  int b = idx / (N * C);
  out[(b * N + n) * C + c] = in[(b * C + c) * N + n];
}

// ---------------- farthest point sampling (one block per batch) -------------
__global__ void fps_kernel(const float* __restrict__ xyz, int N, int npoint,
                           float* __restrict__ new_xyz) {
  int b = blockIdx.x;
  int tid = threadIdx.x;  // blockDim.x == N (<=1024)
  __shared__ float rv[1024];
  __shared__ int ri[1024];
  __shared__ float cen[3];
  float px = xyz[(b * N + tid) * 3 + 0];
  float py = xyz[(b * N + tid) * 3 + 1];
  float pz = xyz[(b * N + tid) * 3 + 2];
  float dist = 1.0e10f;
  int far = 0;
  for (int t = 0; t < npoint; ++t) {
    if (tid == 0) {
      float cx = xyz[(b * N + far) * 3 + 0];
      float cy = xyz[(b * N + far) * 3 + 1];
      float cz = xyz[(b * N + far) * 3 + 2];
      new_xyz[(b * npoint + t) * 3 + 0] = cx;
      new_xyz[(b * npoint + t) * 3 + 1] = cy;
      new_xyz[(b * npoint + t) * 3 + 2] = cz;
      cen[0] = cx; cen[1] = cy; cen[2] = cz;
    }
    __syncthreads();
    float dx = px - cen[0], dy = py - cen[1], dz = pz - cen[2];
    float d = dx * dx + dy * dy + dz * dz;
    dist = fminf(dist, d);
    rv[tid] = dist;
    ri[tid] = tid;
    __syncthreads();
    for (int s = N >> 1; s > 0; s >>= 1) {
      if (tid < s) {
        float v2 = rv[tid + s];
        int i2 = ri[tid + s];
        // argmax with first-index tie-break (matches jnp.argmax)
        if (v2 > rv[tid] || (v2 == rv[tid] && i2 < ri[tid])) {
          rv[tid] = v2; ri[tid] = i2;
        }
      }
      __syncthreads();
    }
    far = ri[0];
    __syncthreads();
  }
}

// ---------------- ball query (one wave per center) --------------------------
__global__ void ball_query_kernel(const float* __restrict__ pts, int N,
                                  const float* __restrict__ centers, int M,
                                  float r2, int* __restrict__ gidx) {
  int g = blockIdx.x;          // b*M + m
  int b = g / M;
  int lane = threadIdx.x;      // blockDim.x == 32 (wave32)
  float cx = centers[g * 3 + 0];
  float cy = centers[g * 3 + 1];
  float cz = centers[g * 3 + 2];
  int* out = gidx + (size_t)g * 64;
  int count = 0;
  int first = -1;
  for (int base = 0; base < N && count < 64; base += 32) {
    int p = base + lane;
    bool inr = false;
    if (p < N) {
      float dx = pts[(b * N + p) * 3 + 0] - cx;
      float dy = pts[(b * N + p) * 3 + 1] - cy;
      float dz = pts[(b * N + p) * 3 + 2] - cz;
      inr = (dx * dx + dy * dy + dz * dz) <= r2;
    }
    unsigned mask = (unsigned)__ballot(inr);
    if (inr) {
      int slot = count + __popc(mask & ((1u << lane) - 1u));
      if (slot < 64) out[slot] = p;
    }
    if (first < 0 && mask) first = base + (__ffs(mask) - 1);
    count += __popc(mask);
  }
  if (count > 64) count = 64;
  int pad = (count == 0) ? (N - 1) : first;
  for (int s = count + lane; s < 64; s += 32) out[s] = pad;
}

// ---------------- fused group MLP: gather -> 3 layers -> maxpool ------------
__global__ void group_mlp_kernel(
    const float* __restrict__ pts, int N,
    const float* __restrict__ feats, int Fch,
    const float* __restrict__ lat, int L,
    const float* __restrict__ centers, const int* __restrict__ gidx, int M,
    const float* __restrict__ P0, const float* __restrict__ s0, const float* __restrict__ b0, int c0,
    const float* __restrict__ P1, const float* __restrict__ s1, const float* __restrict__ b1, int c1,
    const float* __restrict__ P2, const float* __restrict__ s2, const float* __restrict__ b2, int c2,
    float* __restrict__ out, int wA, int wB) {
  int g = blockIdx.x;
  int b = g / M;
  int tid = threadIdx.x;         // 128
  int lane = tid & 31, wave = tid >> 5;
  __shared__ int sg[64];
  __shared__ float sc3[3];
  extern __shared__ float smem[];
  int sA = wA + 2, sB = wB + 2;  // even strides: aligned b64 LDS, no bank clash
  float* bufA = smem;
  float* bufB = smem + 64 * sA;
  if (tid < 64) sg[tid] = gidx[(size_t)g * 64 + tid];
  if (tid < 3) sc3[tid] = centers[g * 3 + tid];
  __syncthreads();
  int K = 3 + Fch + L;
  int kpad = (K + 3) & ~3;
  for (int e = tid; e < 64 * kpad; e += blockDim.x) {
    int r = e / kpad, k = e % kpad;
    float v = 0.0f;
    if (k < 3)             v = pts[(b * N + sg[r]) * 3 + k] - sc3[k];
    else if (k < 3 + Fch)  v = feats[((size_t)b * N + sg[r]) * Fch + (k - 3)];
    else if (k < K)        v = lat[(size_t)g * L + (k - 3 - Fch)];
    bufA[r * sA + k] = v;
  }
  __syncthreads();
  mlp_layer_tile(bufA, sA, bufB, sB, wave * 16, kpad >> 2, c0, P0, s0, b0, nullptr, 0, 0, lane);
  __syncthreads();
  mlp_layer_tile(bufB, sB, bufA, sA, wave * 16, c0 >> 2, c1, P1, s1, b1, nullptr, 0, 0, lane);
  __syncthreads();
  mlp_layer_tile(bufA, sA, bufB, sB, wave * 16, c1 >> 2, c2, P2, s2, b2, nullptr, 0, 0, lane);
  __syncthreads();
  for (int c = tid; c < c2; c += blockDim.x) {     // maxpool over 64 samples
    float mx = bufB[c];
    for (int r = 1; r < 64; ++r) mx = fmaxf(mx, bufB[r * sB + c]);
    out[(size_t)g * c2 + c] = mx;
  }
}

// ---------------- dense row-MLP (enc3 / fold1 / mlp2 stacks) ----------------
__global__ void rows_mlp_kernel(
    const float* __restrict__ X, int R, int cin,
    const float* __restrict__ P0, const float* __restrict__ s0, const float* __restrict__ b0, const float* __restrict__ bias0, int c0,
    const float* __restrict__ P1, const float* __restrict__ s1, const float* __restrict__ b1, const float* __restrict__ bias1, int c1,
    const float* __restrict__ P2, const float* __restrict__ s2, const float* __restrict__ b2, const float* __restrict__ bias2, int c2,
    float* __restrict__ out, int wA, int wB) {
  int tilesPerB = (R + 31) >> 5;
  int b = blockIdx.x / tilesPerB;
  int rowBase = (blockIdx.x % tilesPerB) * 32;
  int tid = threadIdx.x;         // 64
  int lane = tid & 31, wave = tid >> 5;
  extern __shared__ float smem[];
  int sA = wA + 2, sB = wB + 2;
  float* bufA = smem;
  float* bufB = smem + 32 * sA;
  int kpad = (cin + 3) & ~3;
  for (int e = tid; e < 32 * kpad; e += blockDim.x) {
    int r = e / kpad, k = e % kpad;
    int gr = rowBase + r;
    float v = 0.0f;
    if (k < cin && gr < R) v = X[((size_t)b * R + gr) * cin + k];
    bufA[r * sA + k] = v;
  }
  __syncthreads();
  mlp_layer_tile(bufA, sA, bufB, sB, wave * 16, kpad >> 2, c0, P0, s0, b0, bias0, rowBase, R, lane);
  __syncthreads();
  mlp_layer_tile(bufB, sB, bufA, sA, wave * 16, c0 >> 2, c1, P1, s1, b1, bias1, rowBase, R, lane);
  __syncthreads();
  const float* fin = bufA; int sF = sA; int cl = c1;
  if (c2 > 0) {                  // uniform branch (kernel arg)
    mlp_layer_tile(bufA, sA, bufB, sB, wave * 16, c1 >> 2, c2, P2, s2, b2, bias2, rowBase, R, lane);
    fin = bufB; sF = sB; cl = c2;
  }
  __syncthreads();
  for (int e = tid; e < 32 * cl; e += blockDim.x) {
    int r = e / cl, c = e % cl;
    int gr = rowBase + r;
    if (gr < R) out[((size_t)b * R + gr) * cl + c] = fin[r * sF + c];
  }
}

// ---------------- misc small kernels ----------------------------------------
__global__ void maxpool_rows_kernel(const float* __restrict__ in,
                                    float* __restrict__ out, int R, int C) {
  int idx = blockIdx.x * blockDim.x + threadIdx.x;
  if (idx >= BATCH * C) return;
  int b = idx / C, c = idx % C;
  const float* p = in + ((size_t)b * R) * C + c;
  float mx = p[0];
  for (int r = 1; r < R; ++r) mx = fmaxf(mx, p[(size_t)r * C]);
  out[idx] = mx;
}

__global__ void concat_kernel(const float* __restrict__ a, int ca,
                              const float* __restrict__ bsrc, int cb,
                              float* __restrict__ out, int rows) {
  int ct = ca + cb;
  int idx = blockIdx.x * blockDim.x + threadIdx.x;
  if (idx >= rows * ct) return;
  int k = idx % ct, r = idx / ct;
  out[idx] = (k < ca) ? a[(size_t)r * ca + k] : bsrc[(size_t)r * cb + (k - ca)];
}

__global__ void broadcast_code_kernel(const float* __restrict__ code,
                                      float* __restrict__ X) {
  int idx = blockIdx.x * blockDim.x + threadIdx.x;
  const int total = BATCH * 21 * 512;
  if (idx >= total) return;
  int c = idx % 512;
  int b = idx / (21 * 512);
  X[idx] = code[b * 512 + c];
}

// embed[b,j,c] = sum_k x[b,k,c] * G[c,j,k]
__global__ void graph_embed_kernel(const float* __restrict__ x,
                                   const float* __restrict__ G,
                                   float* __restrict__ out, int C) {
  int idx = blockIdx.x * blockDim.x + threadIdx.x;
  if (idx >= BATCH * 21 * C) return;
  int c = idx % C;
  int j = (idx / C) % 21;
  int b = idx / (21 * C);
  const float* g = G + (size_t)c * 441 + j * 21;
  float acc = 0.0f;
  for (int k = 0; k < 21; ++k) acc += x[((size_t)b * 21 + k) * C + c] * g[k];
  out[idx] = acc;
}

// joints = x @ Wr^T + br (+ prev)
__global__ void reg_kernel(const float* __restrict__ x, int C,
                           const float* __restrict__ Wr,
                           const float* __restrict__ br,
                           const float* __restrict__ prev,
                           float* __restrict__ out) {
  int idx = blockIdx.x * blockDim.x + threadIdx.x;
  if (idx >= BATCH * 21 * 3) return;
  int k = idx % 3, bj = idx / 3;
  float acc = br[k];
  const float* xr = x + (size_t)bj * C;
  const float* wr = Wr + (size_t)k * C;
  for (int c = 0; c < C; ++c) acc += xr[c] * wr[c];
  if (prev) acc += prev[idx];
  out[idx] = acc;
}

// ============================================================================
extern "C" void kernel_launch(void* const* d_in, const int* in_sizes, int n_in,
                              void* d_out, int out_size, void* d_ws, size_t ws_size,
                              hipStream_t stream) {
  (void)in_sizes; (void)n_in; (void)out_size; (void)ws_size;
  auto F = [&](int i) { return (const float*)d_in[i]; };

  // ---- parameter pointers: setup_inputs() insertion-order flattening ----
  const float* pc = F(0);
  const float* feat = F(1);
  int p = 2;
  const float *e1W[3], *e1s[3], *e1b[3];
  for (int i = 0; i < 3; ++i) { e1W[i] = F(p++); e1s[i] = F(p++); e1b[i] = F(p++); }
  const float *e2W[3], *e2s[3], *e2b[3];
  for (int i = 0; i < 3; ++i) { e2W[i] = F(p++); e2s[i] = F(p++); e2b[i] = F(p++); }
  const float *e3W[3], *e3s[3], *e3b[3];
  for (int i = 0; i < 3; ++i) { e3W[i] = F(p++); e3s[i] = F(p++); e3b[i] = F(p++); }
  const float *f1W[3], *f1B[3], *f1s[3], *f1b[3];  // fold1: (W, bias, s, b)
  for (int i = 0; i < 3; ++i) { f1W[i] = F(p++); f1B[i] = F(p++); f1s[i] = F(p++); f1b[i] = F(p++); }
  const float* r1W = F(p++); const float* r1b = F(p++);
  const float* graph2 = F(p++);
  const float *m2W[3], *m2s[3], *m2b[3];            // fold2.mlp
  for (int i = 0; i < 3; ++i) { m2W[i] = F(p++); m2s[i] = F(p++); m2b[i] = F(p++); }
  const float *n2W[2], *n2s[2], *n2b[2];            // fold2.mlp2
  for (int i = 0; i < 2; ++i) { n2W[i] = F(p++); n2s[i] = F(p++); n2b[i] = F(p++); }
  const float* r2W = F(p++); const float* r2b = F(p++);
  const float* graph3 = F(p++);
  const float *m3W[3], *m3s[3], *m3b[3];            // fold3.mlp
  for (int i = 0; i < 3; ++i) { m3W[i] = F(p++); m3s[i] = F(p++); m3b[i] = F(p++); }
  const float *n3W[2], *n3s[2], *n3b[2];            // fold3.mlp2
  for (int i = 0; i < 2; ++i) { n3W[i] = F(p++); n3s[i] = F(p++); n3b[i] = F(p++); }
  const float* r3W = F(p++); const float* r3b = F(p++);

  // ---- workspace layout ----
  char* wsb = (char*)d_ws;
  size_t off = 0;
  auto alloc = [&](size_t nbytes) -> void* {
    void* ptr = wsb + off;
    off = (off + nbytes + 255) & ~(size_t)255;
    return ptr;
  };
  float* xyz    = (float*)alloc((size_t)BATCH * 1024 * 3 * 4);
  float* fts    = (float*)alloc((size_t)BATCH * 1024 * 3 * 4);
  float* xyz1   = (float*)alloc((size_t)BATCH * 512 * 3 * 4);
  int*   gidx1  = (int*)  alloc((size_t)BATCH * 512 * 64 * 4);
  float* f1     = (float*)alloc((size_t)BATCH * 512 * 128 * 4);
  float* xyz2   = (float*)alloc((size_t)BATCH * 128 * 3 * 4);
  int*   gidx2  = (int*)  alloc((size_t)BATCH * 128 * 64 * 4);
  float* f2     = (float*)alloc((size_t)BATCH * 128 * 256 * 4);
  float* cat3   = (float*)alloc((size_t)BATCH * 128 * 259 * 4);
  float* e3out  = (float*)alloc((size_t)BATCH * 128 * 512 * 4);
  float* code   = (float*)alloc((size_t)BATCH * 512 * 4);
  float* foldX  = (float*)alloc((size_t)BATCH * 21 * 512 * 4);
  float* fat1   = (float*)alloc((size_t)BATCH * 21 * 128 * 4);
  float* embed  = (float*)alloc((size_t)BATCH * 21 * 256 * 4);
  int*   gidxJ  = (int*)  alloc((size_t)BATCH * 21 * 64 * 4);
  float* pooled = (float*)alloc((size_t)BATCH * 21 * 256 * 4);
  float* locb   = (float*)alloc((size_t)BATCH * 21 * 256 * 4);

  const int TB = 256;
  auto cdiv = [](int a, int b) { return (a + b - 1) / b; };

  // ---- pre-pack all MLP weights into WMMA B-fragment order ----
  auto packW = [&](const float* W, int cin, int cout) -> float* {
    int kpad = (cin + 3) & ~3;
    float* dst = (float*)alloc((size_t)cout * kpad * 4);
    int total = (cout >> 4) * (kpad >> 2) * 64;
    pack_w_kernel<<<cdiv(total, TB), TB, 0, stream>>>(W, cin, cout, dst);
    return dst;
  };
  float* e1P[3] = { packW(e1W[0], 6, 32),   packW(e1W[1], 32, 32),   packW(e1W[2], 32, 128)  };
  float* e2P[3] = { packW(e2W[0], 131, 64), packW(e2W[1], 64, 64),   packW(e2W[2], 64, 256)  };
  float* e3P[3] = { packW(e3W[0], 259, 128),packW(e3W[1], 128, 128), packW(e3W[2], 128, 512) };
  float* f1P[3] = { packW(f1W[0], 512, 256),packW(f1W[1], 256, 256), packW(f1W[2], 256, 128) };
  float* m2P[3] = { packW(m2W[0], 259, 256),packW(m2W[1], 256, 256), packW(m2W[2], 256, 256) };
  float* n2P[2] = { packW(n2W[0], 256, 256),packW(n2W[1], 256, 256) };
  float* m3P[3] = { packW(m3W[0], 387, 256),packW(m3W[1], 256, 256), packW(m3W[2], 256, 256) };
  float* n3P[2] = { packW(n3W[0], 256, 256),packW(n3W[1], 256, 256) };

  float* outF = (float*)d_out;
  float* j1 = outF;           // (32,21,3) flat == reshape(-1,63)
  float* j2 = outF + 2016;
  float* j3 = outF + 4032;

  auto ldsG = [](int wA, int wB) { return (size_t)(64 * (wA + 2) + 64 * (wB + 2)) * 4; };
  auto ldsR = [](int wA, int wB) { return (size_t)(32 * (wA + 2) + 32 * (wB + 2)) * 4; };

  // ---- transpose inputs ----
  transpose_bcn_kernel<<<cdiv(BATCH * 3 * 1024, TB), TB, 0, stream>>>(pc, xyz, 3, 1024);
  transpose_bcn_kernel<<<cdiv(BATCH * 3 * 1024, TB), TB, 0, stream>>>(feat, fts, 3, 1024);

  // ---- SA1: FPS 512, r=0.1, ns=64, MLP [6->32->32->128] ----
  fps_kernel<<<BATCH, 1024, 0, stream>>>(xyz, 1024, 512, xyz1);
  ball_query_kernel<<<BATCH * 512, 32, 0, stream>>>(xyz, 1024, xyz1, 512, 0.01f, gidx1);
  group_mlp_kernel<<<BATCH * 512, 128, ldsG(32, 128), stream>>>(
      xyz, 1024, fts, 3, nullptr, 0, xyz1, gidx1, 512,
      e1P[0], e1s[0], e1b[0], 32,
      e1P[1], e1s[1], e1b[1], 32,
      e1P[2], e1s[2], e1b[2], 128,
      f1, 32, 128);

  // ---- SA2: FPS 128, r=0.3, ns=64, MLP [131->64->64->256] ----
  fps_kernel<<<BATCH, 512, 0, stream>>>(xyz1, 512, 128, xyz2);
  ball_query_kernel<<<BATCH * 128, 32, 0, stream>>>(xyz1, 512, xyz2, 128, 0.09f, gidx2);
  group_mlp_kernel<<<BATCH * 128, 128, ldsG(132, 256), stream>>>(
      xyz1, 512, f1, 128, nullptr, 0, xyz2, gidx2, 128,
      e2P[0], e2s[0], e2b[0], 64,
      e2P[1], e2s[1], e2b[1], 64,
      e2P[2], e2s[2], e2b[2], 256,
      f2, 132, 256);

  // ---- enc3: [259->128->128->512] + maxpool over 128 points -> code ----
  concat_kernel<<<cdiv(BATCH * 128 * 259, TB), TB, 0, stream>>>(xyz2, 3, f2, 256, cat3, BATCH * 128);
  rows_mlp_kernel<<<BATCH * 4, 64, ldsR(260, 512), stream>>>(
      cat3, 128, 259,
      e3P[0], e3s[0], e3b[0], nullptr, 128,
      e3P[1], e3s[1], e3b[1], nullptr, 128,
      e3P[2], e3s[2], e3b[2], nullptr, 512,
      e3out, 260, 512);
  maxpool_rows_kernel<<<cdiv(BATCH * 512, TB), TB, 0, stream>>>(e3out, code, 128, 512);

  // ---- fold1: broadcast code over 21 joints, [512->256->256->128] w/ bias ----
  broadcast_code_kernel<<<cdiv(BATCH * 21 * 512, TB), TB, 0, stream>>>(code, foldX);
  rows_mlp_kernel<<<BATCH, 64, ldsR(512, 256), stream>>>(
      foldX, 21, 512,
      f1P[0], f1s[0], f1b[0], f1B[0], 256,
      f1P[1], f1s[1], f1b[1], f1B[1], 256,
      f1P[2], f1s[2], f1b[2], f1B[2], 128,
      fat1, 512, 256);
  reg_kernel<<<cdiv(BATCH * 21 * 3, TB), TB, 0, stream>>>(fat1, 128, r1W, r1b, nullptr, j1);

  // ---- fold2 mapping: embed + ball query joints1 + MLP [259->256^3] + mlp2 ----
  graph_embed_kernel<<<cdiv(BATCH * 21 * 128, TB), TB, 0, stream>>>(fat1, graph2, embed, 128);
  ball_query_kernel<<<BATCH * 21, 32, 0, stream>>>(xyz1, 512, j1, 21, 0.09f, gidxJ);
  group_mlp_kernel<<<BATCH * 21, 128, ldsG(260, 256), stream>>>(
      xyz1, 512, f1, 128, embed, 128, j1, gidxJ, 21,
      m2P[0], m2s[0], m2b[0], 256,
      m2P[1], m2s[1], m2b[1], 256,
      m2P[2], m2s[2], m2b[2], 256,
      pooled, 260, 256);
  rows_mlp_kernel<<<BATCH, 64, ldsR(256, 256), stream>>>(
      pooled, 21, 256,
      n2P[0], n2s[0], n2b[0], nullptr, 256,
      n2P[1], n2s[1], n2b[1], nullptr, 256,
      nullptr, nullptr, nullptr, nullptr, 0,
      locb, 256, 256);
  reg_kernel<<<cdiv(BATCH * 21 * 3, TB), TB, 0, stream>>>(locb, 256, r2W, r2b, j1, j2);

  // ---- fold3 mapping: embed + ball query joints2 + MLP [387->256^3] + mlp2 ----
  graph_embed_kernel<<<cdiv(BATCH * 21 * 256, TB), TB, 0, stream>>>(locb, graph3, embed, 256);
  ball_query_kernel<<<BATCH * 21, 32, 0, stream>>>(xyz1, 512, j2, 21, 0.09f, gidxJ);
  group_mlp_kernel<<<BATCH * 21, 128, ldsG(388, 256), stream>>>(
      xyz1, 512, f1, 128, embed, 256, j2, gidxJ, 21,
      m3P[0], m3s[0], m3b[0], 256,
      m3P[1], m3s[1], m3b[1], 256,
      m3P[2], m3s[2], m3b[2], 256,
      pooled, 388, 256);
  rows_mlp_kernel<<<BATCH, 64, ldsR(256, 256), stream>>>(
      pooled, 21, 256,
      n3P[0], n3s[0], n3b[0], nullptr, 256,
      n3P[1], n3s[1], n3b[1], nullptr, 256,
      nullptr, nullptr, nullptr, nullptr, 0,
      locb, 256, 256);
  reg_kernel<<<cdiv(BATCH * 21 * 3, TB), TB, 0, stream>>>(locb, 256, r3W, r3b, j2, j3);
}